// PredictorModel_5248450035985
// MI455X (gfx1250) — compile-verified
//
#include <hip/hip_runtime.h>
#include <math.h>

// ---------------- problem constants ----------------
#define T_N   2048
#define R_N   64
#define NTX_N 64
#define NE_N  16
#define RC    (R_N * NTX_N)      // 4096 columns
#define P_N   6400
#define B_N   2
#define PI_F  3.14159265358979323846f

typedef __attribute__((ext_vector_type(2))) float v2f;
typedef __attribute__((ext_vector_type(8))) float v8f;

// ---------------- complex helpers ----------------
__device__ __forceinline__ float2 cmul(float2 a, float2 b) {
  return make_float2(a.x * b.x - a.y * b.y, a.x * b.y + a.y * b.x);
}
__device__ __forceinline__ float2 cmulc(float2 a, float2 b) { // conj(a)*b
  return make_float2(a.x * b.x + a.y * b.y, a.x * b.y - a.y * b.x);
}
__device__ __forceinline__ float2 crecip(float2 a) {
  float d = 1.0f / (a.x * a.x + a.y * a.y);
  return make_float2(a.x * d, -a.y * d);
}

// ---------------- WMMA wrappers (CDNA5 V_WMMA_F32_16X16X4_F32) ----------------
// 8-arg form: (neg_a, A, neg_b, B, c_mod, C, reuse_a, reuse_b); neg_a/neg_b must
// be 0 for F32 per ISA NEG table, so imaginary-part subtraction negates the
// fragment in VALU instead (compiler lowers it to v_xor sign-flips).
__device__ __forceinline__ v8f wmma_f32(v2f a, v2f b, v8f c) {
  return __builtin_amdgcn_wmma_f32_16x16x4_f32(false, a, false, b, (short)0, c,
                                               false, false);
}

// A-matrix 16x4 fragment from complex row-major LDS (ldc = complex leading dim).
// Lanes 0-15: M=lane, K={k0,k0+1}; lanes 16-31: M=lane-16, K={k0+2,k0+3}.
__device__ __forceinline__ v2f frag_a(const float* M, int row0, int k0, int ldc,
                                      int comp) {
  int l = threadIdx.x & 31;
  int m = row0 + (l & 15);
  int k = k0 + ((l >> 4) << 1);
  v2f a;
  a.x = M[(m * ldc + k) * 2 + comp];
  a.y = M[(m * ldc + k + 1) * 2 + comp];
  return a;
}
// B-matrix 4x16 fragment: VGPR v holds rows K={k0+v, k0+2+v} split by lane half.
__device__ __forceinline__ v2f frag_b(const float* M, int k0, int col0, int ldc,
                                      int comp) {
  int l = threadIdx.x & 31;
  int n = col0 + (l & 15);
  int kb = k0 + ((l >> 4) << 1);
  v2f b;
  b.x = M[(kb * ldc + n) * 2 + comp];
  b.y = M[((kb + 1) * ldc + n) * 2 + comp];
  return b;
}
// C/D 16x16 f32: VGPR v -> row (v | v+8 by lane half), N = lane&15.
__device__ __forceinline__ void store_cd(float* M, int row0, int col0, int ldc,
                                         v8f cr, v8f ci) {
  int l = threadIdx.x & 31;
  int n = col0 + (l & 15);
  int mb = row0 + ((l >> 4) << 3);
#pragma unroll
  for (int v = 0; v < 8; ++v) {
    M[((mb + v) * ldc + n) * 2] = cr[v];
    M[((mb + v) * ldc + n) * 2 + 1] = ci[v];
  }
}

// ---------------- Stockham radix-2 FFT in LDS (N = 2048) ----------------
__device__ __forceinline__ float2* stockham2048(float2* bufA, float2* bufB,
                                                float sgn) {
  float2* src = bufA;
  float2* dst = bufB;
  int l = 1024;
  for (int s = 0; s < 11; ++s) {
    int m = 1 << s;
    __syncthreads();
    for (int p = (int)threadIdx.x; p < 1024; p += (int)blockDim.x) {
      int j = p >> s;
      int k = p & (m - 1);
      float ang = sgn * PI_F * (float)j / (float)l;
      float sw, cw;
      __sincosf(ang, &sw, &cw);
      float2 u = src[k + m * j];
      float2 v = src[k + m * (j + l)];
      float2 su = make_float2(u.x + v.x, u.y + v.y);
      float2 di = make_float2(u.x - v.x, u.y - v.y);
      dst[k + m * (2 * j)] = su;
      dst[k + m * (2 * j + 1)] =
          make_float2(di.x * cw - di.y * sw, di.x * sw + di.y * cw);
    }
    float2* t = src;
    src = dst;
    dst = t;
    l >>= 1;
  }
  __syncthreads();
  return src;
}

// Forward FFT of real input column c: X[t][c] = FFT_t(datas[t][c])
__global__ void __launch_bounds__(256) fft_fwd_kernel(
    const float* __restrict__ in, float2* __restrict__ X) {
  __shared__ float2 bufA[T_N], bufB[T_N];
  const int c = blockIdx.x;
  for (int t = threadIdx.x; t < T_N; t += blockDim.x)
    bufA[t] = make_float2(in[(size_t)t * RC + c], 0.0f);
  float2* res = stockham2048(bufA, bufB, -1.0f);
  for (int t = threadIdx.x; t < T_N; t += blockDim.x)
    X[(size_t)t * RC + c] = res[t];
}

// Inverse FFT of W (rows 0..1024 stored, upper half implicitly zero from the
// Hilbert multiplier) -> iq transposed to [c][t] for fast beamform gathers.
__global__ void __launch_bounds__(256) ifft_kernel(const float2* __restrict__ W,
                                                   float2* __restrict__ iq) {
  __shared__ float2 bufA[T_N], bufB[T_N];
  const int c = blockIdx.x;
  for (int t = threadIdx.x; t < T_N; t += blockDim.x)
    bufA[t] = (t <= T_N / 2) ? W[(size_t)t * RC + c] : make_float2(0.f, 0.f);
  float2* res = stockham2048(bufA, bufB, 1.0f);
  const float scl = 1.0f / (float)T_N;
  for (int t = threadIdx.x; t < T_N; t += blockDim.x) {
    float2 v = res[t];
    iq[(size_t)c * T_N + t] = make_float2(v.x * scl, v.y * scl);
  }
}

// ---------------- fused per-frequency-pair encode/decode ----------------
// Block handles (f, nf=(2048-f)%2048). Computes H, Hinv (Gauss-Jordan on
// [A|H^H], A = H^H H + 0.1 I), Y = X@H (WMMA), Hermitian-symmetrize
// (== fft(real(ifft(.)))), Z = Xe@Hinv (WMMA), symmetrize + Hilbert weight,
// writes W[f] (rows > T/2 are zero and never stored). The 64KB X staging runs
// as GLOBAL_LOAD_ASYNC_TO_LDS_B128 (ASYNCcnt) fully overlapped with the solve.
__global__ void __launch_bounds__(256) encode_decode_kernel(
    const float2* __restrict__ X, const float* __restrict__ delays,
    const float* __restrict__ weights, float2* __restrict__ W) {
  extern __shared__ float sm[];
  float* sX = sm;            // [2][64][64] cplx (reused as sZ), 16384 f
  float* sH = sm + 16384;    // [2][64][16] cplx, 4096 f
  float* sHi = sm + 20480;   // [2][16][64] cplx, 4096 f
  float* sY = sm + 24576;    // [2][64][16] cplx, 4096 f
  float* sXe = sm + 28672;   // [2][64][16] cplx, 4096 f
  float* sM = sm + 24576;    // [2][16][80] cplx solve scratch (dead before sY)
  __shared__ float2 sFac[32];

  const int tid = threadIdx.x;
  const int f = blockIdx.x;                // 0..1024
  const int nf = (T_N - f) & (T_N - 1);

  // ---- async DMA of both X rows into LDS; no VGPR staging, overlaps solve ----
  {
    const char* xbase = (const char*)X;
    // low 32 bits of a generic LDS pointer == LDS offset (aperture layout)
    const unsigned lds0 = (unsigned)(uintptr_t)sX;
#pragma unroll
    for (int pass = 0; pass < 16; ++pass) {
      int i4 = tid + pass * 256;           // float4 index (4096 total)
      int q = (i4 >> 11) ? nf : f;         // 2048 float4 per frequency row
      const char* gaddr =
          xbase + (size_t)q * (RC * 2 * 4) + (size_t)(i4 & 2047) * 16;
      unsigned laddr = lds0 + (unsigned)i4 * 16;
      asm volatile("global_load_async_to_lds_b128 %0, %1, off"
                   :
                   : "v"(laddr), "v"(gaddr)
                   : "memory");
    }
  }

  // ---- H[fi][tx][e] = w[e][tx] * exp(-2πi * fn(q) * delay[e][tx]) ----
  for (int i = tid; i < 2 * NTX_N * NE_N; i += blockDim.x) {
    int q = (i >> 10) ? nf : f;
    int id = i & 1023;
    int tx = id >> 4, e = id & 15;
    float fn = (float)((q < T_N / 2) ? q : q - T_N) / (float)T_N;
    float ang = -2.0f * PI_F * fn * delays[e * NTX_N + tx];
    float sw, cw;
    __sincosf(ang, &sw, &cw);
    float wt = weights[e * NTX_N + tx];
    sH[i * 2] = wt * cw;
    sH[i * 2 + 1] = wt * sw;
  }
  __syncthreads();

  // ---- build augmented [A | H^H], A = H^H H + TIK*I ----
  for (int i = tid; i < 2 * 16 * 80; i += blockDim.x) {
    int fi = i / 1280, rm = i % 1280, rr = rm / 80, cc = rm % 80;
    const float* Hf = sH + fi * 2048;
    float2 val;
    if (cc < 16) {
      float2 acc = make_float2((rr == cc) ? 0.1f : 0.0f, 0.0f);
      for (int tx = 0; tx < NTX_N; ++tx) {
        float2 hi = make_float2(Hf[(tx * 16 + rr) * 2], Hf[(tx * 16 + rr) * 2 + 1]);
        float2 hj = make_float2(Hf[(tx * 16 + cc) * 2], Hf[(tx * 16 + cc) * 2 + 1]);
        float2 t = cmulc(hi, hj);
        acc.x += t.x;
        acc.y += t.y;
      }
      val = acc;
    } else {
      int tx = cc - 16;  // H^H[rr][tx] = conj(H[tx][rr])
      val = make_float2(Hf[(tx * 16 + rr) * 2], -Hf[(tx * 16 + rr) * 2 + 1]);
    }
    sM[i * 2] = val.x;
    sM[i * 2 + 1] = val.y;
  }
  __syncthreads();

  // ---- Gauss-Jordan (Hermitian PD after Tikhonov: no pivoting) ----
  for (int p = 0; p < 16; ++p) {
    float2 piv0 = make_float2(sM[(p * 80 + p) * 2], sM[(p * 80 + p) * 2 + 1]);
    float2 piv1 = make_float2(sM[(1280 + p * 80 + p) * 2],
                              sM[(1280 + p * 80 + p) * 2 + 1]);
    __syncthreads();
    for (int i = tid; i < 160; i += blockDim.x) {
      int fi = i / 80, cc = i % 80;
      float2 inv = crecip(fi ? piv1 : piv0);
      int o = (fi * 1280 + p * 80 + cc) * 2;
      float2 nv = cmul(make_float2(sM[o], sM[o + 1]), inv);
      sM[o] = nv.x;
      sM[o + 1] = nv.y;
    }
    __syncthreads();
    if (tid < 32) {
      int o = ((tid >> 4) * 1280 + (tid & 15) * 80 + p) * 2;
      sFac[tid] = make_float2(sM[o], sM[o + 1]);
    }
    __syncthreads();
    for (int i = tid; i < 2560; i += blockDim.x) {
      int fi = i / 1280, rm = i % 1280, rr = rm / 80, cc = rm % 80;
      if (rr == p) continue;
      float2 fac = sFac[fi * 16 + rr];
      int op = (fi * 1280 + p * 80 + cc) * 2;
      float2 u = make_float2(sM[op], sM[op + 1]);
      float2 t = cmul(fac, u);
      int o = i * 2;
      sM[o] -= t.x;
      sM[o + 1] -= t.y;
    }
    __syncthreads();
  }
  // ---- Hinv[fi][e][tx] = solution columns ----
  for (int i = tid; i < 2 * NE_N * NTX_N; i += blockDim.x) {
    int fi = i >> 10, rm = i & 1023, e = rm >> 6, tx = rm & 63;
    int o = (fi * 1280 + e * 80 + 16 + tx) * 2;
    sHi[i * 2] = sM[o];
    sHi[i * 2 + 1] = sM[o + 1];
  }

  // ---- drain the async X DMA (per-wave), then make it block-visible ----
  asm volatile("s_wait_asynccnt 0x0" ::: "memory");
  __syncthreads();

  // ---- matmul 1: Y[fi] = X[fi] (64x64) @ H[fi] (64x16), complex via WMMA ----
  const int wave = tid >> 5;
  const int fi_w = wave >> 2;        // 0 -> f, 1 -> nf
  const int rt = (wave & 3) * 16;    // output row tile
  {
    v8f cr = {0, 0, 0, 0, 0, 0, 0, 0}, ci = {0, 0, 0, 0, 0, 0, 0, 0};
    const float* A = sX + fi_w * 8192;
    const float* Bm = sH + fi_w * 2048;
    for (int k0 = 0; k0 < 64; k0 += 4) {
      v2f ar = frag_a(A, rt, k0, 64, 0);
      v2f ai = frag_a(A, rt, k0, 64, 1);
      v2f br = frag_b(Bm, k0, 0, 16, 0);
      v2f bi = frag_b(Bm, k0, 0, 16, 1);
      v2f nai = -ai;
      cr = wmma_f32(ar, br, cr);
      cr = wmma_f32(nai, bi, cr);
      ci = wmma_f32(ar, bi, ci);
      ci = wmma_f32(ai, br, ci);
    }
    store_cd(sY + fi_w * 2048, rt, 0, 16, cr, ci);
  }
  __syncthreads();

  // ---- Xe[fi] = 0.5*(Y[fi] + conj(Y[1-fi]))  (== fft(real(ifft(Y)))) ----
  for (int i = tid; i < 2048; i += blockDim.x) {
    int fi = i >> 10, rm = i & 1023;
    float yr = sY[i * 2], yi = sY[i * 2 + 1];
    float zr = sY[((1 - fi) * 1024 + rm) * 2];
    float zi = sY[((1 - fi) * 1024 + rm) * 2 + 1];
    sXe[i * 2] = 0.5f * (yr + zr);
    sXe[i * 2 + 1] = 0.5f * (yi - zi);
  }
  __syncthreads();

  // ---- matmul 2: Z[fi] = Xe[fi] (64x16) @ Hinv[fi] (16x64) ----
  float* sZ = sX;  // X is dead; reuse its LDS
  {
    const float* A = sXe + fi_w * 2048;
    const float* Bm = sHi + fi_w * 2048;
    for (int ct = 0; ct < 4; ++ct) {
      v8f zr = {0, 0, 0, 0, 0, 0, 0, 0}, zi = {0, 0, 0, 0, 0, 0, 0, 0};
      for (int k0 = 0; k0 < 16; k0 += 4) {
        v2f ar = frag_a(A, rt, k0, 16, 0);
        v2f ai = frag_a(A, rt, k0, 16, 1);
        v2f br = frag_b(Bm, k0, ct * 16, 64, 0);
        v2f bi = frag_b(Bm, k0, ct * 16, 64, 1);
        v2f nai = -ai;
        zr = wmma_f32(ar, br, zr);
        zr = wmma_f32(nai, bi, zr);
        zi = wmma_f32(ar, bi, zi);
        zi = wmma_f32(ai, br, zi);
      }
      store_cd(sZ + fi_w * 8192, rt, ct * 16, 64, zr, zi);
    }
  }
  __syncthreads();

  // ---- W[f] = h(f) * 0.5*(Z[f] + conj(Z[nf])); upper spectrum stays zero ----
  const float h = (f == 0 || f == T_N / 2) ? 1.0f : 2.0f;
  for (int i = tid; i < RC; i += blockDim.x) {
    float z0r = sZ[i * 2], z0i = sZ[i * 2 + 1];
    float z1r = sZ[8192 + i * 2], z1i = sZ[8192 + i * 2 + 1];
    W[(size_t)f * RC + i] =
        make_float2(h * 0.5f * (z0r + z1r), h * 0.5f * (z0i - z1i));
  }
}

// ---------------- delay-and-sum beamformer (iq resident in 192MB L2) --------
__global__ void __launch_bounds__(256) beamform_kernel(
    const float2* __restrict__ iq, const float* __restrict__ bf,
    float* __restrict__ img) {
  const int p = blockIdx.x;
  const int tid = threadIdx.x;
  __shared__ float sbf[R_N];
  __shared__ float2 red[256];
  if (tid < R_N) sbf[tid] = bf[(size_t)tid * P_N + p];
  __syncthreads();
  const float inv_dr = 25.0e6f / 1540.0e3f;  // fs / c  (1/DR), R0 = 0
  float ax = 0.0f, ay = 0.0f;
  for (int j = tid; j < RC; j += blockDim.x) {
    int r = j >> 6, tx = j & 63;
    float idx = (sbf[r] + sbf[tx]) * inv_dr;
    int i0 = (int)floorf(idx);
    i0 = i0 < 0 ? 0 : (i0 > T_N - 2 ? T_N - 2 : i0);
    float frac = idx - (float)i0;
    const float2* col = iq + (size_t)j * T_N;
    float2 g0 = col[i0];
    float2 g1 = col[i0 + 1];
    if (idx >= 0.0f && idx <= (float)(T_N - 1)) {
      ax += g0.x + frac * (g1.x - g0.x);
      ay += g0.y + frac * (g1.y - g0.y);
    }
  }
  red[tid] = make_float2(ax, ay);
  __syncthreads();
  for (int s = 128; s > 0; s >>= 1) {
    if (tid < s) {
      red[tid].x += red[tid + s].x;
      red[tid].y += red[tid + s].y;
    }
    __syncthreads();
  }
  if (tid == 0)
    img[p] = sqrtf(red[0].x * red[0].x + red[0].y * red[0].y) + 1e-15f;
}

// ---------------- per-batch dB normalization ----------------
__global__ void __launch_bounds__(256) dbnorm_kernel(
    const float* __restrict__ img, float* __restrict__ out) {
  const int b = blockIdx.x;
  const int tid = threadIdx.x;
  __shared__ float red[256];
  float m = 0.0f;
  for (int i = tid; i < P_N; i += blockDim.x) m = fmaxf(m, img[b * P_N + i]);
  red[tid] = m;
  __syncthreads();
  for (int s = 128; s > 0; s >>= 1) {
    if (tid < s) red[tid] = fmaxf(red[tid], red[tid + s]);
    __syncthreads();
  }
  float mx = red[0];
  for (int i = tid; i < P_N; i += blockDim.x) {
    float db = 20.0f * log10f(img[b * P_N + i] / mx);
    out[b * P_N + i] = fminf(fmaxf(db, -60.0f), 0.0f);
  }
}

// ---------------- host driver ----------------
extern "C" void kernel_launch(void* const* d_in, const int* in_sizes, int n_in,
                              void* d_out, int out_size, void* d_ws,
                              size_t ws_size, hipStream_t stream) {
  (void)in_sizes; (void)n_in; (void)out_size; (void)ws_size;
  const float* datas = (const float*)d_in[0];    // [B,T,R,NTX]
  const float* delays = (const float*)d_in[1];   // [NE,NTX]
  const float* weights = (const float*)d_in[2];  // [NE,NTX]
  const float* bf = (const float*)d_in[3];       // [R,P]
  float* out = (float*)d_out;                    // [B,NZ,NX]

  char* ws = (char*)d_ws;
  const size_t xbytes = (size_t)T_N * RC * sizeof(float2);           // 64 MB
  const size_t wbytes = (size_t)(T_N / 2 + 1) * RC * sizeof(float2); // 32 MB
  float2* Xbuf = (float2*)ws;                  // X spectrum, reused as iq[c][t]
  float2* Wbuf = (float2*)(ws + xbytes);       // half-spectrum W
  float* img = (float*)(ws + xbytes + wbytes); // [B][P]

  for (int b = 0; b < B_N; ++b) {
    fft_fwd_kernel<<<RC, 256, 0, stream>>>(datas + (size_t)b * T_N * RC, Xbuf);
    encode_decode_kernel<<<T_N / 2 + 1, 256, 131072, stream>>>(
        Xbuf, delays, weights, Wbuf);
    ifft_kernel<<<RC, 256, 0, stream>>>(Wbuf, Xbuf);  // Xbuf now holds iq
    beamform_kernel<<<P_N, 256, 0, stream>>>(Xbuf, bf, img + (size_t)b * P_N);
  }
  dbnorm_kernel<<<B_N, 256, 0, stream>>>(img, out);
}